// Ex2Vec_59923383714074
// MI455X (gfx1250) — compile-verified
//
#include <hip/hip_runtime.h>

typedef __attribute__((ext_vector_type(2))) float v2f;
typedef __attribute__((ext_vector_type(8))) float v8f;
typedef __attribute__((ext_vector_type(4))) int   v4i;

#define D_DIM   64
#define H_DIM   128
#define P_TOT   1000
#define P_TILE  64
#define ROWSTR  68          // padded LDS row stride (68 % 64 == 4 -> conflict-free frag loads)
#define EPS_F   1e-12f
#define LOG2E_F 1.4426950408889634f

// ---- CDNA5 async global->LDS copy path (guarded: falls back to VGPR staging) ----
#if defined(__AMDGCN__) && __has_builtin(__builtin_amdgcn_global_load_async_to_lds_b128) && __has_builtin(__builtin_amdgcn_s_wait_asynccnt)
#define USE_ASYNC_LDS 1
typedef v4i __attribute__((address_space(1)))* gbl_v4i_p;   // global (AS1)
typedef v4i __attribute__((address_space(3)))* lds_v4i_p;   // LDS    (AS3)
#else
#define USE_ASYNC_LDS 0
#endif

__device__ __forceinline__ void copy16_to_lds(const float* __restrict__ g, float* l) {
#if USE_ASYNC_LDS
    __builtin_amdgcn_global_load_async_to_lds_b128(
        (gbl_v4i_p)(void*)g,      // addrspacecast generic->global, const dropped
        (lds_v4i_p)(void*)l,      // addrspacecast generic->LDS (proper offset conversion)
        /*offset=*/0, /*cpol=*/0);
#else
    const float4 v = *(const float4*)g;
    l[0] = v.x; l[1] = v.y; l[2] = v.z; l[3] = v.w;
#endif
}

__device__ __forceinline__ void wait_lds_copies() {
#if USE_ASYNC_LDS
    __builtin_amdgcn_s_wait_asynccnt(0);
#endif
}

__global__ __launch_bounds__(256) void ex2vec_kernel(
    const int*   __restrict__ user_index,          // [B]
    const int*   __restrict__ pred_item_indices,   // [B,P]
    const int*   __restrict__ history_item_indices,// [B,H]
    const float* __restrict__ history_timedeltas,  // [B,H]
    const float* __restrict__ history_weights,     // [B,H]
    const float* __restrict__ embedding_user,      // [N_USERS+1, D]
    const float* __restrict__ embedding_item,      // [N_ITEMS+1, D]
    const float* __restrict__ user_lamb,           // [N_USERS+1]
    const float* __restrict__ user_bias,           // [N_USERS+1]
    const float* __restrict__ item_bias,           // [N_ITEMS+1]
    const float* __restrict__ s_global_lamb,
    const float* __restrict__ s_alpha,
    const float* __restrict__ s_beta,
    const float* __restrict__ s_gamma,
    const float* __restrict__ s_cutoff,
    const float* __restrict__ s_smooth,
    const float* __restrict__ s_force,
    float*       __restrict__ out)                 // [B,P]
{
    __shared__ float h_s[H_DIM * ROWSTR];      // history tile   (34.8 KB)
    __shared__ float p_s[P_TILE * ROWSTR];     // candidate tile (17.4 KB)
    __shared__ float hh_s[H_DIM];              // |h|^2 per row
    __shared__ float td_s[H_DIM];              // (t+cutoff)^-1/2 * w
    __shared__ float pp_s[P_TILE];             // |p|^2 per col
    __shared__ float dui_s[P_TILE];            // dist(u,p) per col
    __shared__ float res_s[P_TILE];            // td-weighted kern sum per col
    __shared__ float u_s[D_DIM];
    __shared__ int   pidx_s[P_TILE];

    const int tid   = threadIdx.x;
    const int lane  = tid & 31;
    const int wave  = tid >> 5;                // 8 waves (wave32)
    const int b     = blockIdx.y;
    const int pbase = blockIdx.x * P_TILE;

    const int uidx = user_index[b];

    // ---- stage u-vector, candidate indices, zero accumulators ----
    if (tid < D_DIM) u_s[tid] = embedding_user[(size_t)uidx * D_DIM + tid];
    if (tid >= D_DIM && tid < D_DIM + P_TILE) {
        const int c  = tid - D_DIM;
        const int pc = pbase + c;
        pidx_s[c] = (pc < P_TOT) ? pred_item_indices[(size_t)b * P_TOT + pc] : 0;
        res_s[c]  = 0.0f;
    }
    __syncthreads();

    // ---- gather history + candidate rows into LDS (async DMA when available) ----
    for (int i = tid; i < H_DIM * 16; i += 256) {
        const int r = i >> 4, c4 = i & 15;
        const int hidx = history_item_indices[(size_t)b * H_DIM + r];
        copy16_to_lds(embedding_item + (size_t)hidx * D_DIM + c4 * 4,
                      &h_s[r * ROWSTR + c4 * 4]);
    }
    for (int i = tid; i < P_TILE * 16; i += 256) {
        const int r = i >> 4, c4 = i & 15;
        copy16_to_lds(embedding_item + (size_t)pidx_s[r] * D_DIM + c4 * 4,
                      &p_s[r * ROWSTR + c4 * 4]);
    }
    wait_lds_copies();
    __syncthreads();

    const float cutoff = *s_cutoff;
    const float smooth = *s_smooth;
    const float force  = *s_force;
    const float fs     = force * smooth;
    const float inv_denom = 1.0f + __expf(fs - smooth);   // 1/sigmoid(smooth - fs)

    // ---- row norms, time decay, user-candidate distance ----
    if (tid < H_DIM) {
        float s = 0.0f;
        #pragma unroll 8
        for (int d = 0; d < D_DIM; ++d) { const float x = h_s[tid * ROWSTR + d]; s += x * x; }
        hh_s[tid] = s;
        td_s[tid] = rsqrtf(history_timedeltas[(size_t)b * H_DIM + tid] + cutoff)
                    * history_weights[(size_t)b * H_DIM + tid];
    } else if (tid < H_DIM + P_TILE) {
        const int c = tid - H_DIM;
        float sp = 0.0f, sd = 0.0f;
        #pragma unroll 8
        for (int d = 0; d < D_DIM; ++d) {
            const float x = p_s[c * ROWSTR + d];
            sp += x * x;
            const float dd = u_s[d] - x;
            sd += dd * dd;
        }
        pp_s[c]  = sp;
        dui_s[c] = __builtin_amdgcn_sqrtf(sd + EPS_F);
    }
    __syncthreads();

    // ---- Gram tile via fp32 WMMA: wave w owns H-strip rows [16w, 16w+16) ----
    // f32 A 16x4 layout: lanes 0-15 -> M=lane, K pair (0,1); lanes 16-31 -> K pair (2,3)
    const int hm    = wave * 16;
    const int lrow  = lane & 15;
    const int khalf = (lane >> 4) << 1;          // 0 or 2
    const int mhi   = (lane >> 4) << 3;          // C rows r / r+8 per lane half

    const float* ap = &h_s[(hm + lrow) * ROWSTR + khalf];

    for (int nt = 0; nt < P_TILE / 16; ++nt) {
        const float* bp = &p_s[(nt * 16 + lrow) * ROWSTR + khalf];
        v8f acc = {0.f, 0.f, 0.f, 0.f, 0.f, 0.f, 0.f, 0.f};
        #pragma unroll
        for (int kk = 0; kk < D_DIM / 4; ++kk) {
            v2f a, bb;
            a.x  = ap[kk * 4];     a.y  = ap[kk * 4 + 1];
            bb.x = bp[kk * 4];     bb.y = bp[kk * 4 + 1];
            acc = __builtin_amdgcn_wmma_f32_16x16x4_f32(
                      false, a, false, bb, (short)0, acc, false, false);
        }

        // fused epilogue: sq -> dist -> kern via raw HW transcendentals,
        // weighted reduce over M (H rows)
        const int   nloc = nt * 16 + lrow;
        const float ppv  = pp_s[nloc];
        float partial = 0.0f;
        #pragma unroll
        for (int r = 0; r < 8; ++r) {
            const int   m    = hm + r + mhi;
            const float gram = acc[r];
            const float sq   = hh_s[m] + ppv - 2.0f * gram;
            const float dist = __builtin_amdgcn_sqrtf(fmaxf(sq, 0.0f) + EPS_F);
            const float x    = smooth * __builtin_amdgcn_rcpf(1.0f + dist) - fs;
            // sigmoid(x)/denom = inv_denom / (1 + 2^(-x*log2e))
            const float e    = __builtin_amdgcn_exp2f(-x * LOG2E_F);
            const float kern = inv_denom * __builtin_amdgcn_rcpf(1.0f + e);
            partial += td_s[m] * kern;
        }
        partial += __shfl_xor(partial, 16, 32);  // combine lane halves (M 0-7 + 8-15)
        if (lane < 16) atomicAdd(&res_s[nloc], partial);   // ds_add_f32 across 8 waves
    }
    __syncthreads();

    // ---- final score ----
    if (tid < P_TILE) {
        const int pc = pbase + tid;
        if (pc < P_TOT) {
            const float lamb = *s_global_lamb + user_lamb[uidx];
            const float resv = lamb * res_s[tid];
            const float o    = fmaxf(0.0f, dui_s[tid] - resv);
            const float I    = (*s_alpha) * o + (*s_beta) * o * o + (*s_gamma)
                               + user_bias[uidx] + item_bias[pidx_s[tid]];
            out[(size_t)b * P_TOT + pc] = I;
        }
    }
}

extern "C" void kernel_launch(void* const* d_in, const int* in_sizes, int n_in,
                              void* d_out, int out_size, void* d_ws, size_t ws_size,
                              hipStream_t stream) {
    const int B = in_sizes[0];
    dim3 grid((P_TOT + P_TILE - 1) / P_TILE, B);
    ex2vec_kernel<<<grid, 256, 0, stream>>>(
        (const int*)  d_in[0],   // user_index
        (const int*)  d_in[1],   // pred_item_indices
        (const int*)  d_in[2],   // history_item_indices
        (const float*)d_in[3],   // history_timedeltas
        (const float*)d_in[4],   // history_weights
        (const float*)d_in[5],   // embedding_user
        (const float*)d_in[6],   // embedding_item
        (const float*)d_in[7],   // user_lamb
        (const float*)d_in[8],   // user_bias
        (const float*)d_in[9],   // item_bias
        (const float*)d_in[10],  // global_lamb
        (const float*)d_in[11],  // alpha
        (const float*)d_in[12],  // beta
        (const float*)d_in[13],  // gamma
        (const float*)d_in[14],  // cutoff
        (const float*)d_in[15],  // smooth
        (const float*)d_in[16],  // force
        (float*)d_out);
}